// GCN_10333691314775
// MI455X (gfx1250) — compile-verified
//
#include <hip/hip_runtime.h>
#include <hip/hip_bf16.h>

typedef __attribute__((ext_vector_type(2))) float v2f;
typedef __attribute__((ext_vector_type(8))) float v8f;

#define GCN_N 50000

// ---------------------------------------------------------------------------
// deg[dst] += 1 over edges (self-loop "+1" folded into gcn_dinv)
// ---------------------------------------------------------------------------
__global__ void gcn_degree(const int* __restrict__ dst, float* __restrict__ deg, int E) {
  int i = blockIdx.x * blockDim.x + threadIdx.x;
  if (i < E) atomicAdd(&deg[dst[i]], 1.0f);
}

__global__ void gcn_dinv(float* __restrict__ deg, int n) {
  int i = blockIdx.x * blockDim.x + threadIdx.x;
  if (i < n) deg[i] = rsqrtf(deg[i] + 1.0f);  // deg+1 >= 1 always (self loop)
}

// ---------------------------------------------------------------------------
// C[M,N] = A[M,K] @ W[K,N], f32, via V_WMMA_F32_16X16X4_F32.
// K, N are template constants so all B/A/C addresses inside the unroll window
// fold into 24-bit instruction immediates (one pointer bump per window).
// One wave computes a 16x32 tile (two accumulators share one A fragment);
// 2 waves per 64-thread block cover 64 columns.
// A fragment (16x4, ISA 7.12.2): lanes 0-15 hold K={0,1}, lanes 16-31 K={2,3}.
// B fragment (4x16): row K striped across lanes; same K-half split as A.
// C/D: VGPR i -> row m0+i (lanes 0-15) / m0+8+i (lanes 16-31).
// Requires M%16==0, N%64==0, K%4==0 (true here: 50000 / 512,256 / 256,64).
// ---------------------------------------------------------------------------
template <int K, int N>
__global__ void gcn_gemm_wmma(const float* __restrict__ A, const float* __restrict__ W,
                              float* __restrict__ C) {
  const int lane = threadIdx.x & 31;
  const int wave = threadIdx.x >> 5;
  const int m0 = blockIdx.x << 4;
  const int n0 = ((blockIdx.y << 1) + wave) << 5;   // 32 columns per wave

  const int mrow = m0 + (lane & 15);
  const int koff = (lane >> 4) << 1;      // 0 for lanes 0-15, 2 for lanes 16-31
  const int ncol = n0 + (lane & 15);

  const float* ap = A + (size_t)mrow * K + koff;    // 8B-aligned (koff even)
  const float* bp = W + (size_t)koff * N + ncol;

  v8f acc0 = {};
  v8f acc1 = {};
#pragma unroll 8
  for (int k = 0; k < K / 4; ++k) {
    v2f a = *(const v2f*)(ap + 4 * k);              // A[mrow][4k+koff .. +1]
    v2f b0, b1;
    b0[0] = bp[4 * k * N];                          // W[4k+koff  ][ncol]
    b0[1] = bp[4 * k * N + N];                      // W[4k+koff+1][ncol]
    b1[0] = bp[4 * k * N + 16];                     // W[4k+koff  ][ncol+16]
    b1[1] = bp[4 * k * N + N + 16];                 // W[4k+koff+1][ncol+16]
    acc0 = __builtin_amdgcn_wmma_f32_16x16x4_f32(false, a, false, b0, (short)0,
                                                 acc0, false, false);
    acc1 = __builtin_amdgcn_wmma_f32_16x16x4_f32(false, a, false, b1, (short)0,
                                                 acc1, false, false);
  }

  const int mbase = m0 + ((lane >> 4) << 3);        // +8 for upper half-wave
  float* crow = C + (size_t)mbase * N + ncol;
#pragma unroll
  for (int i = 0; i < 8; ++i) {
    crow[i * N] = acc0[i];
    crow[i * N + 16] = acc1[i];
  }
}

// ---------------------------------------------------------------------------
// Edge scatter: acc[dst] += h[src] * dinv[src]*dinv[dst].  One wave per edge;
// lane strides by 32 -> coalesced cacheline loads and L2-resident f32 atomics.
// ---------------------------------------------------------------------------
template <int F>
__global__ void gcn_scatter(const float* __restrict__ h, const int* __restrict__ src,
                            const int* __restrict__ dst, const float* __restrict__ dinv,
                            float* __restrict__ acc, int E) {
  int w = blockIdx.x * (blockDim.x >> 5) + (threadIdx.x >> 5);
  int lane = threadIdx.x & 31;
  if (w >= E) return;
  int s = src[w], d = dst[w];
  float norm = dinv[s] * dinv[d];
  const float* hp = h + (size_t)s * F;
  float* op = acc + (size_t)d * F;
#pragma unroll
  for (int f = lane; f < F; f += 32)
    atomicAdd(op + f, hp[f] * norm);
}

// ---------------------------------------------------------------------------
// Epilogue: out = acc + h*dinv^2 (self-loop message) + bias; optional ReLU.
// Safe to run in place (out == acc).
// ---------------------------------------------------------------------------
template <int F, bool RELU>
__global__ void gcn_finalize(const float* __restrict__ acc, const float* __restrict__ h,
                             const float* __restrict__ dinv, const float* __restrict__ bias,
                             float* __restrict__ out, int n) {
  long long i = (long long)blockIdx.x * blockDim.x + threadIdx.x;
  if (i >= (long long)n * F) return;
  int node = (int)(i / F);   // F is a power of two -> shifts
  int f = (int)(i % F);
  float di = dinv[node];
  float v = acc[i] + h[i] * (di * di) + bias[f];
  if (RELU) v = fmaxf(v, 0.0f);
  out[i] = v;
}

// ---------------------------------------------------------------------------
extern "C" void kernel_launch(void* const* d_in, const int* in_sizes, int n_in,
                              void* d_out, int out_size, void* d_ws, size_t ws_size,
                              hipStream_t stream) {
  (void)n_in; (void)out_size; (void)ws_size;

  const float* x  = (const float*)d_in[0];   // [50000,512]
  const int*   ei = (const int*)d_in[1];     // [2,E]
  const float* W1 = (const float*)d_in[2];   // [512,256]
  const float* b1 = (const float*)d_in[3];
  const float* W2 = (const float*)d_in[4];   // [256,256]
  const float* b2 = (const float*)d_in[5];
  const float* W3 = (const float*)d_in[6];   // [256,64]
  const float* b3 = (const float*)d_in[7];

  const int E = in_sizes[1] / 2;
  const int* src = ei;
  const int* dst = ei + E;

  // Workspace: dinv | bufT (GEMM out) | bufH (aggregated)  — each big buffer
  // is 51.2MB; both fit in the 192MB L2 together, keeping gather/scatter hot.
  char* ws = (char*)d_ws;
  size_t off0 = ((size_t)GCN_N * 4 + 255) & ~(size_t)255;
  size_t big  = (size_t)GCN_N * 256 * 4;
  float* dinv = (float*)ws;
  float* bufT = (float*)(ws + off0);
  float* bufH = (float*)(ws + off0 + big);
  float* outp = (float*)d_out;               // [50000,64]

  // --- normalization ---
  hipMemsetAsync(dinv, 0, (size_t)GCN_N * 4, stream);
  gcn_degree<<<(E + 255) / 256, 256, 0, stream>>>(dst, dinv, E);
  gcn_dinv<<<(GCN_N + 255) / 256, 256, 0, stream>>>(dinv, GCN_N);

  const dim3 gblk(64);  // 2 waves per block, 64 output columns per block

  // --- layer 1: T = X @ W1 (K=512,N=256); aggregate; +b1, ReLU -> bufH ---
  gcn_gemm_wmma<512, 256><<<dim3(GCN_N / 16, 256 / 64), gblk, 0, stream>>>(x, W1, bufT);
  hipMemsetAsync(bufH, 0, big, stream);
  gcn_scatter<256><<<(E + 7) / 8, 256, 0, stream>>>(bufT, src, dst, dinv, bufH, E);
  gcn_finalize<256, true><<<(GCN_N * 256 + 255) / 256, 256, 0, stream>>>(
      bufH, bufT, dinv, b1, bufH, GCN_N);

  // --- layer 2: T = H @ W2 (K=256,N=256); aggregate; +b2, ReLU -> bufH ---
  gcn_gemm_wmma<256, 256><<<dim3(GCN_N / 16, 256 / 64), gblk, 0, stream>>>(bufH, W2, bufT);
  hipMemsetAsync(bufH, 0, big, stream);   // ordered after GEMM read of bufH
  gcn_scatter<256><<<(E + 7) / 8, 256, 0, stream>>>(bufT, src, dst, dinv, bufH, E);
  gcn_finalize<256, true><<<(GCN_N * 256 + 255) / 256, 256, 0, stream>>>(
      bufH, bufT, dinv, b2, bufH, GCN_N);

  // --- layer 3: T = H @ W3 (K=256,N=64); aggregate; +b3 -> d_out ---
  gcn_gemm_wmma<256, 64><<<dim3(GCN_N / 16, 64 / 64), gblk, 0, stream>>>(bufH, W3, bufT);
  hipMemsetAsync(outp, 0, (size_t)GCN_N * 64 * 4, stream);
  gcn_scatter<64><<<(E + 7) / 8, 256, 0, stream>>>(bufT, src, dst, dinv, outp, E);
  gcn_finalize<64, false><<<(GCN_N * 64 + 255) / 256, 256, 0, stream>>>(
      outp, bufT, dinv, b3, outp, GCN_N);
}